// GCN_Encoder_66529043415295
// MI455X (gfx1250) — compile-verified
//
#include <hip/hip_runtime.h>
#include <cstddef>

typedef __attribute__((ext_vector_type(2))) float v2f;
typedef __attribute__((ext_vector_type(8))) float v8f;

// ---------------------------------------------------------------------------
// Degree accumulation: deg[dst] += ew   (self-loop +2.0 folded into rsqrt pass)
// ---------------------------------------------------------------------------
__global__ __launch_bounds__(256) void degree_kernel(
    const long long* __restrict__ dst, const float* __restrict__ ew,
    float* __restrict__ deg, int E) {
  int i = blockIdx.x * blockDim.x + threadIdx.x;
  if (i < E) atomicAdd(&deg[dst[i]], ew[i]);
}

__global__ __launch_bounds__(256) void rsqrt_kernel(float* __restrict__ d, int N) {
  int i = blockIdx.x * blockDim.x + threadIdx.x;
  if (i < N) d[i] = rsqrtf(d[i] + 2.0f);  // deg always > 0 (self-loop weight 2)
}

// ---------------------------------------------------------------------------
// FP32 WMMA GEMM: C[N x FOUT] = A[N x 128] * B[128 x FOUT]
// One wave -> 16 rows x FOUT cols, using V_WMMA_F32_16X16X4_F32.
// A frag (16x4): lanes 0-15 hold K=+0/+1, lanes 16-31 hold K=+2/+3.
// B frag (4x16): mirrored; D: 8 VGPRs, lane halves = M 0-7 / 8-15.
// ---------------------------------------------------------------------------
template <int FOUT>
__global__ __launch_bounds__(128) void gemm_wmma(
    const float* __restrict__ A, const float* __restrict__ B,
    float* __restrict__ C, int N) {
  constexpr int K = 128;
  constexpr int NT = FOUT / 16;

  const int lane = threadIdx.x & 31;
  const int wave = threadIdx.x >> 5;
  const int row0 = (blockIdx.x * 4 + wave) * 16;
  if (row0 >= N) return;  // wave-uniform exit: EXEC stays all-ones for WMMA

  const int m  = lane & 15;
  const int kk = (lane >> 4) << 1;  // 0 for lanes 0-15, 2 for lanes 16-31

  v8f acc[NT] = {};
  const float* arow = A + (size_t)(row0 + m) * K + kk;

  for (int k = 0; k < K; k += 4) {
    v2f a;
    a.x = arow[k];
    a.y = arow[k + 1];
#pragma unroll
    for (int t = 0; t < NT; ++t) {
      const float* bp = B + (size_t)(k + kk) * FOUT + t * 16 + m;
      v2f b;
      b.x = bp[0];
      b.y = bp[FOUT];
      acc[t] = __builtin_amdgcn_wmma_f32_16x16x4_f32(
          /*neg_a=*/false, a, /*neg_b=*/false, b,
          /*c_mod=*/(short)0, acc[t], /*reuse_a=*/false, /*reuse_b=*/false);
    }
  }

  const int mrow = (lane >> 4) << 3;  // 0 or 8
#pragma unroll
  for (int t = 0; t < NT; ++t) {
#pragma unroll
    for (int r = 0; r < 8; ++r) {
      C[(size_t)(row0 + mrow + r) * FOUT + t * 16 + m] = acc[t][r];
    }
  }
}

// ---------------------------------------------------------------------------
// Edge aggregation: out[dst] += dinv[src]*w*dinv[dst] * h[src]
// One wave per edge: coalesced gather of h[src] + L2-resident float atomics.
// ---------------------------------------------------------------------------
template <int F>
__global__ __launch_bounds__(256) void agg_edges(
    const float* __restrict__ h, const long long* __restrict__ src,
    const long long* __restrict__ dst, const float* __restrict__ ew,
    const float* __restrict__ dinv, float* __restrict__ out, int E) {
  constexpr int VE = F / 32;  // floats per lane (4 @F=128, 2 @F=64)
  const int e = blockIdx.x * (blockDim.x >> 5) + (threadIdx.x >> 5);
  if (e >= E) return;
  const int lane = threadIdx.x & 31;

  const long long s = src[e];
  const long long d = dst[e];
  const float norm = dinv[s] * ew[e] * dinv[d];

  const float* hp = h + (size_t)s * F + lane * VE;
  float* op = out + (size_t)d * F + lane * VE;

  if constexpr (VE == 4) {
    float4 v = *(const float4*)hp;
    atomicAdd(op + 0, norm * v.x);
    atomicAdd(op + 1, norm * v.y);
    atomicAdd(op + 2, norm * v.z);
    atomicAdd(op + 3, norm * v.w);
  } else {
    float2 v = *(const float2*)hp;
    atomicAdd(op + 0, norm * v.x);
    atomicAdd(op + 1, norm * v.y);
  }
}

// ---------------------------------------------------------------------------
// Self-loop + bias + activation (in place on agg):
//   agg = act( agg + 2*dinv^2*hself + bias )
// ---------------------------------------------------------------------------
template <int F, bool RELU>
__global__ __launch_bounds__(256) void post_kernel(
    float* __restrict__ agg, const float* __restrict__ hself,
    const float* __restrict__ dinv, const float* __restrict__ bias, int total) {
  int idx = blockIdx.x * blockDim.x + threadIdx.x;
  if (idx >= total) return;
  const int i = idx / F;
  const int f = idx & (F - 1);
  const float di = dinv[i];
  float v = agg[idx] + 2.0f * di * di * hself[idx] + bias[f];
  if constexpr (RELU) {
    v = fmaxf(v, 0.0f);
  } else {
    v = 1.0f / (1.0f + __expf(-v));  // sigmoid
  }
  agg[idx] = v;
}

// ---------------------------------------------------------------------------
extern "C" void kernel_launch(void* const* d_in, const int* in_sizes, int n_in,
                              void* d_out, int out_size, void* d_ws, size_t ws_size,
                              hipStream_t stream) {
  const float*     x  = (const float*)d_in[0];      // [N,128]
  const long long* ed = (const long long*)d_in[1];  // [2,E] int64
  const float*     ew = (const float*)d_in[2];      // [E]
  const float*     W0 = (const float*)d_in[3];      // [128,128]
  const float*     b0 = (const float*)d_in[4];      // [128]
  const float*     W1 = (const float*)d_in[5];      // [128,64]
  const float*     b1 = (const float*)d_in[6];      // [64]

  const int E = in_sizes[2];
  const int N = in_sizes[0] / 128;
  const long long* src = ed;
  const long long* dst = ed + E;

  float* dinv = (float*)d_ws;                 // [N]
  float* bufA = dinv + N;                     // [N,128]  h1, later h2 ([N,64])
  float* bufB = bufA + (size_t)N * 128;       // [N,128]  agg1, then relu'd h1
  float* out  = (float*)d_out;                // [N,64]

  const int eBlocks    = (E + 255) / 256;
  const int nBlocks    = (N + 255) / 256;
  const int waveBlocks = (E + 7) / 8;                   // 8 waves (edges) / block
  const int gemmBlocks = ((N + 15) / 16 + 3) / 4;       // 4 waves (16-row tiles) / block

  // dinv = rsqrt(deg + 2)
  hipMemsetAsync(dinv, 0, (size_t)N * sizeof(float), stream);
  degree_kernel<<<eBlocks, 256, 0, stream>>>(dst, ew, dinv, E);
  rsqrt_kernel<<<nBlocks, 256, 0, stream>>>(dinv, N);

  // Layer 1: h1 = x @ W0 ; agg ; relu(agg + self + b0) -> bufB
  gemm_wmma<128><<<gemmBlocks, 128, 0, stream>>>(x, W0, bufA, N);
  hipMemsetAsync(bufB, 0, (size_t)N * 128 * sizeof(float), stream);
  agg_edges<128><<<waveBlocks, 256, 0, stream>>>(bufA, src, dst, ew, dinv, bufB, E);
  post_kernel<128, true><<<(N * 128 + 255) / 256, 256, 0, stream>>>(bufB, bufA, dinv, b0, N * 128);

  // Layer 2: h2 = bufB @ W1 -> bufA ; agg into d_out ; sigmoid(+self+b1)
  gemm_wmma<64><<<gemmBlocks, 128, 0, stream>>>(bufB, W1, bufA, N);
  hipMemsetAsync(out, 0, (size_t)N * 64 * sizeof(float), stream);
  agg_edges<64><<<waveBlocks, 256, 0, stream>>>(bufA, src, dst, ew, dinv, out, E);
  post_kernel<64, false><<<(N * 64 + 255) / 256, 256, 0, stream>>>(out, bufA, dinv, b1, N * 64);
}